// SimpleAttn_3839700762908
// MI455X (gfx1250) — compile-verified
//
#include <hip/hip_runtime.h>
#include <hip/hip_bf16.h>

typedef __attribute__((ext_vector_type(2))) float v2f;
typedef __attribute__((ext_vector_type(4))) float v4f;
typedef __attribute__((ext_vector_type(8))) float v8f;
typedef __attribute__((ext_vector_type(4))) unsigned int v4u;
typedef __attribute__((ext_vector_type(4))) int v4i;
typedef __attribute__((ext_vector_type(8))) int v8i;

#define B_   8
#define XL_  128
#define QL_  16
#define VL_  256
#define D_   256

#define APAD 260   // LDS row pitch (floats) -> conflict-free strided frag reads
#define SPAD 17    // pitch for S^T [16][17]
#define NTHREADS 512   // 16 wave32

#if defined(__has_builtin)
#if __has_builtin(__builtin_amdgcn_tensor_load_to_lds)
#define HAVE_TDM 1
#endif
#endif
#ifndef HAVE_TDM
#define HAVE_TDM 0
#endif

__device__ __forceinline__ v8f wmma_f32(v2f a, v2f b, v8f c) {
  // D = A(16x4 f32) * B(4x16 f32) + C(16x16 f32)
  return __builtin_amdgcn_wmma_f32_16x16x4_f32(false, a, false, b, (short)0, c,
                                               false, false);
}

__device__ __forceinline__ v8f zero8() {
  v8f z = {0.f, 0.f, 0.f, 0.f, 0.f, 0.f, 0.f, 0.f};
  return z;
}

#if HAVE_TDM
__device__ __forceinline__ unsigned lds_off(const void* p) {
  return (unsigned)(size_t)p;   // generic LDS pointer: low 32 bits = LDS byte offset
}

// TDM: DMA a 16-row x 256-f32 tile (row pitch 256 f32 in memory) into LDS with
// hardware padding of 4 DWORDs every 256 DWORDs -> LDS pitch = APAD(260) floats.
__device__ __forceinline__ void tdm_load_16x256(const float* gsrc, unsigned lds_addr) {
  unsigned long long ga = (unsigned long long)(size_t)gsrc;
  v4u g0 = { 1u,                                   // count=1 (valid descriptor)
             lds_addr,                             // lds_addr
             (unsigned)(ga & 0xffffffffu),         // global_addr[31:0]
             (unsigned)((ga >> 32) & 0x01ffffffu)  // global_addr[56:32]
               | (2u << 30) };                     // type = 2 ("image")
  v8i g1 = { (int)((2u << 16)     // data_size = 4 bytes
               | (1u << 20)       // pad_enable
               | (7u << 22)       // pad_interval: 256 DWORDs (one 256-f32 row)
               | (3u << 25)),     // pad_amount:   4 DWORDs  -> pitch 260 floats
             (int)(256u << 16),   // tensor_dim0[15:0] = 256  (bits 79:48)
             (int)(16u << 16),    // tensor_dim0 hi = 0 | tensor_dim1[15:0] = 16
             (int)(256u << 16),   // tensor_dim1 hi = 0 | tile_dim0 = 256
             16,                  // tile_dim1 = 16 | tile_dim2 = 0
             256,                 // tensor_dim0_stride[31:0] = 256 elements
             0,                   // stride0 hi | tensor_dim1_stride lo (unused, 2D)
             0 };
  v4i gz = {0, 0, 0, 0};          // groups 2/3: 2-D tensor, unused
#if defined(__clang_major__) && __clang_major__ >= 23
  v8i gz8 = {0, 0, 0, 0, 0, 0, 0, 0};
  __builtin_amdgcn_tensor_load_to_lds(g0, g1, gz, gz, gz8, 0);
#else
  __builtin_amdgcn_tensor_load_to_lds(g0, g1, gz, gz, 0);
#endif
}
#endif  // HAVE_TDM

__global__ __launch_bounds__(NTHREADS, 1)
void fused_simple_attn(const float* __restrict__ q, const float* __restrict__ k,
                       const float* __restrict__ v,
                       const unsigned char* __restrict__ mask,
                       const float* __restrict__ qw_w, const float* __restrict__ qw_b,
                       const float* __restrict__ kw_w, const float* __restrict__ kw_b,
                       const float* __restrict__ vw_w, const float* __restrict__ vw_b,
                       float* __restrict__ out) {
  __shared__ float s_stage[2][16 * APAD];  // double-buffered tiles; kp written in-place
  __shared__ float s_qp[16 * APAD];        // qp (plain [q][d], padded rows)
  __shared__ float s_S[16 * SPAD];         // S^T block : [v][q]
  __shared__ float s_ta[256];
  __shared__ float s_attn[256];
  __shared__ float s_out[256];
  __shared__ float s_red[2];               // [0]=sum ta^2, [1]=sum exp
  __shared__ int   s_cnt;                  // # unmasked

  const int bx   = blockIdx.x;             // b*XL + x  (0..1023)
  const int tid  = threadIdx.x;
  const int wave = tid >> 5;               // 0..15 -> col-block owner
  const int lane = tid & 31;
  const int n16  = lane & 15;
  const int sel  = (lane >> 4) << 1;       // 0 (lanes 0-15) or 2 (lanes 16-31)
  const int d0   = wave * 16;              // this wave's 16-wide output col block
  const int row8 = (lane < 16) ? 0 : 8;    // C/D upper-half row offset

  // ---- manual (VGPR) staging of a 16x256 tile: used for q, and as TDM fallback ----
  auto stage16x256 = [&](float* dst, const float* __restrict__ src) {
    const v4f* p = (const v4f*)(src + wave * 256 + lane * 8);
    v4f x0 = p[0];
    v4f x1 = p[1];
    float* dd = dst + wave * APAD + lane * 8;
    *(v4f*)(dd)     = x0;
    *(v4f*)(dd + 4) = x1;
  };

  // ---- preload this wave's weight B-fragments (16 out-dims x K=256) into VGPRs ----
  // B[kk][n] = W[(d0+n)*256 + kk]  (W row-major [out][in], computing x @ W^T)
  auto preload_w = [&](const float* __restrict__ W, v2f* wf) {
    const v2f* wp = (const v2f*)W;
    const int base = (d0 + n16) * 128 + (sel >> 1);   // in v2f units
#pragma unroll
    for (int t = 0; t < 64; ++t) wf[t] = wp[base + 2 * t];
  };

  // ---- one 16x16 output col-block: A from LDS tile, B from preloaded regs ----
  auto proj_colblock = [&](const float* abuf, const v2f* wf) -> v8f {
    v8f c = zero8();
    const float* arow = abuf + n16 * APAD + sel;
#pragma unroll
    for (int t = 0; t < 64; ++t) {
      v2f a = *(const v2f*)(arow + 4 * t);
      c = wmma_f32(a, wf[t], c);
    }
    return c;
  };

  v2f wf[64];   // 128 VGPRs of weight fragments, reused per phase

  // =========================== Phase Q: qp = q W_q^T + b_q ===========================
  preload_w(qw_w, wf);
  stage16x256(&s_stage[0][0], q + (size_t)bx * QL_ * D_);
  __syncthreads();
  {
    v8f c = proj_colblock(&s_stage[0][0], wf);
    float bias = qw_b[d0 + n16];
#pragma unroll
    for (int r = 0; r < 8; ++r)
      s_qp[(r + row8) * APAD + d0 + n16] = c[r] + bias;
  }
  __syncthreads();   // qp visible; all reads of s_stage[0] complete

  // ======================= Phase K: kp blocks + S^T + ta (max_q) =====================
  preload_w(kw_w, wf);
  {
    const float* kbase = k + (size_t)bx * VL_ * D_;
    float kbias = kw_b[d0 + n16];
#if HAVE_TDM
    if (wave == 0) tdm_load_16x256(kbase, lds_off(&s_stage[0][0]));  // block 0 -> buf0
#endif
    if (tid < 16 * SPAD) s_S[tid] = 0.f;

    for (int vb = 0; vb < 16; ++vb) {
      const int cur = vb & 1;
#if HAVE_TDM
      if (wave == 0) {
        if (vb < 15) {   // issue next DMA, then require only the OLDER one done
          tdm_load_16x256(kbase + (size_t)(vb + 1) * 16 * D_,
                          lds_off(&s_stage[cur ^ 1][0]));
          __builtin_amdgcn_s_wait_tensorcnt((short)1);
        } else {
          __builtin_amdgcn_s_wait_tensorcnt((short)0);
        }
      }
#else
      stage16x256(&s_stage[cur][0], kbase + (size_t)vb * 16 * D_);
#endif
      __syncthreads();                       // BAR-A: tile vb visible in buf[cur]

      v8f c = proj_colblock(&s_stage[cur][0], wf);   // kp block (this wave's 16 d)
      __syncthreads();                       // BAR-B: raw-k reads of buf[cur] done
#pragma unroll
      for (int r = 0; r < 8; ++r)            // kp block written in place over raw k
        s_stage[cur][(r + row8) * APAD + d0 + n16] = c[r] + kbias;
      __syncthreads();                       // BAR-C: kp block visible

      // S^T[v][q] += sum_d kp[v][d] * qp[q][d]; each wave takes a K-slice of 16
      {
        v8f s = zero8();
        const float* arow = &s_stage[cur][n16 * APAD + sel];
        const float* brow = &s_qp[n16 * APAD + sel];
#pragma unroll
        for (int j = 0; j < 4; ++j) {
          const int kk = wave * 16 + 4 * j;
          v2f a = *(const v2f*)(arow + kk);
          v2f b = *(const v2f*)(brow + kk);
          s = wmma_f32(a, b, s);
        }
#pragma unroll
        for (int r = 0; r < 8; ++r)
          atomicAdd(&s_S[(r + row8) * SPAD + n16], s[r]);
      }
      __syncthreads();                       // BAR-D: S^T complete

      if (tid < 16) {                        // ta[v] = max over q; re-zero S^T
        float m = s_S[tid * SPAD];
#pragma unroll
        for (int qq = 1; qq < 16; ++qq) m = fmaxf(m, s_S[tid * SPAD + qq]);
        s_ta[vb * 16 + tid] = m;
#pragma unroll
        for (int t2 = 0; t2 < SPAD; ++t2) s_S[tid * SPAD + t2] = 0.f;
      }
    }
  }
  __syncthreads();

  // Kick off V phase DMA of block 0 so it overlaps the softmax math below.
  preload_w(vw_w, wf);
  const float* vbase = v + (size_t)bx * VL_ * D_;
#if HAVE_TDM
  if (wave == 0) tdm_load_16x256(vbase, lds_off(&s_stage[0][0]));
#endif

  // ===================== normalize / mask / softmax -> s_attn ========================
  if (tid == 0) { s_red[0] = 0.f; s_red[1] = 0.f; s_cnt = 0; }
  __syncthreads();
  {
    float tav = (tid < 256) ? s_ta[tid] : 0.f;
    float sq = tav * tav;
#pragma unroll
    for (int off = 16; off > 0; off >>= 1) sq += __shfl_down(sq, off, 32);
    if (tid < 256 && lane == 0) atomicAdd(&s_red[0], sq);
    __syncthreads();

    float nrm = sqrtf(s_red[0]);             // L2 norm BEFORE masking (as reference)
    int mk = (tid < 256) ? (int)mask[(size_t)bx * 256 + tid] : 1;
    float val = mk ? -10.0f : (tav / nrm);
    float e = (tid < 256) ? __expf(val) : 0.f;  // values bounded; no max-shift needed
    float es = e;
    int cnt = (tid < 256 && !mk) ? 1 : 0;
#pragma unroll
    for (int off = 16; off > 0; off >>= 1) {
      es += __shfl_down(es, off, 32);
      cnt += __shfl_down(cnt, off, 32);
    }
    if (tid < 256 && lane == 0) { atomicAdd(&s_red[1], es); atomicAdd(&s_cnt, cnt); }
    __syncthreads();

    if (tid < 256) {
      s_attn[tid] = (s_cnt == 0) ? 0.f : (e / s_red[1]);
      s_out[tid] = 0.f;
    }
  }
  __syncthreads();

  // ============ Phase V: vp blocks consumed in-register: out += attn * vp ============
  {
    float vbias = vw_b[d0 + n16];
    for (int vb = 0; vb < 16; ++vb) {
      const int cur = vb & 1;
#if HAVE_TDM
      if (wave == 0) {
        if (vb < 15) {
          tdm_load_16x256(vbase + (size_t)(vb + 1) * 16 * D_,
                          lds_off(&s_stage[cur ^ 1][0]));
          __builtin_amdgcn_s_wait_tensorcnt((short)1);
        } else {
          __builtin_amdgcn_s_wait_tensorcnt((short)0);
        }
      }
#else
      stage16x256(&s_stage[cur][0], vbase + (size_t)vb * 16 * D_);
#endif
      __syncthreads();                       // tile vb visible

      v8f c = proj_colblock(&s_stage[cur][0], wf);
      float acc = 0.f;
#pragma unroll
      for (int r = 0; r < 8; ++r)
        acc += s_attn[vb * 16 + r + row8] * (c[r] + vbias);
      atomicAdd(&s_out[d0 + n16], acc);
      __syncthreads();                       // reads of buf[cur] done before reuse
    }
  }
  if (tid < 256) out[(size_t)bx * 256 + tid] = s_out[tid];
}

extern "C" void kernel_launch(void* const* d_in, const int* in_sizes, int n_in,
                              void* d_out, int out_size, void* d_ws, size_t ws_size,
                              hipStream_t stream) {
  const float* q  = (const float*)d_in[0];
  const float* k  = (const float*)d_in[1];
  const float* v  = (const float*)d_in[2];
  const unsigned char* mask = (const unsigned char*)d_in[3];  // jax bool -> 1 byte
  const float* qw_w = (const float*)d_in[4];
  const float* qw_b = (const float*)d_in[5];
  const float* kw_w = (const float*)d_in[6];
  const float* kw_b = (const float*)d_in[7];
  const float* vw_w = (const float*)d_in[8];
  const float* vw_b = (const float*)d_in[9];
  float* out = (float*)d_out;

  dim3 grid(B_ * XL_);        // one workgroup per (b, x)
  dim3 block(NTHREADS);
  hipLaunchKernelGGL(fused_simple_attn, grid, block, 0, stream,
                     q, k, v, mask, qw_w, qw_b, kw_w, kw_b, vw_w, vw_b, out);
}